// DiffKS_73504070304242
// MI455X (gfx1250) — compile-verified
//
#include <hip/hip_runtime.h>
#include <math.h>

typedef float v2f __attribute__((ext_vector_type(2)));
typedef float v8f __attribute__((ext_vector_type(8)));
typedef int   v4i __attribute__((ext_vector_type(4)));

#define T_SAMPLES 44100
#define N_FRAMES_ 100
#define NUM_COEFF_ 6
#define BURST 2048
#define EXC_ORDER_ 5
#define CHUNK 101        // min feedback lag (delay >= 100 -> lag >= 101)

#if defined(__has_builtin)
#if __has_builtin(__builtin_amdgcn_global_load_async_to_lds_b128) && __has_builtin(__builtin_amdgcn_s_wait_asynccnt)
#define KS_HAVE_ASYNC 1
#endif
#endif

static __device__ __forceinline__ float sigm(float v) { return 1.0f / (1.0f + expf(-v)); }

static __device__ __forceinline__ void exc_interp(int t, const float* ec, float a[EXC_ORDER_]) {
  float p = (float)t * (99.0f / 2047.0f);
  int j0 = (int)floorf(p);
  j0 = j0 < 0 ? 0 : (j0 > 98 ? 98 : j0);
  float w = p - (float)j0;
#pragma unroll
  for (int k = 0; k < EXC_ORDER_; ++k) {
    float lo = ec[j0 * EXC_ORDER_ + k];
    float hi = ec[(j0 + 1) * EXC_ORDER_ + k];
    a[k] = lo + (hi - lo) * w;
  }
}

// ---------------------------------------------------------------------------
// Kernel 1: per-sample tap precompute. b(16x6) = W(16x4) x frames(4x16) via
// V_WMMA_F32_16X16X4_F32 per 16-sample row block (one block per wave), then
// per-sample vals[t] = {v0..v6, z_l} written to workspace.
// ---------------------------------------------------------------------------
__global__ __launch_bounds__(256) void ks_coeff_wmma(
    const float* __restrict__ delay_frames,
    const float* __restrict__ raw_coeff,
    const float* __restrict__ raw_gain,
    float* __restrict__ vals) {
  __shared__ float d_s[8][16][8];
  const float fscale = 99.0f / 44099.0f;

  int wave = threadIdx.x >> 5;
  int lane = threadIdx.x & 31;
  int bstart = blockIdx.x * 128 + wave * 16;

  // Uniform base frame index for this 16-sample block (frames advance ~0.0022/sample)
  float p0 = (float)bstart * fscale;
  int base = (int)floorf(p0);
  base = base < 0 ? 0 : (base > 96 ? 96 : base);

  float gain = sigm(raw_gain[0]) * 0.1f + 0.9f;

  // A matrix (16x4 f32): lane 0-15 rows, VGPR0/1 = K{0,1}; lanes 16-31 same rows, K{2,3}
  int row = lane & 15;
  int t = bstart + row;
  float pos = (float)t * fscale;
  int i0 = (int)floorf(pos);
  i0 = i0 < 0 ? 0 : (i0 > 98 ? 98 : i0);
  float w = pos - (float)i0;
  int k0 = i0 - base;  // 0..2 within block
  bool tv = (t < T_SAMPLES);
  int kA = (lane < 16) ? 0 : 2;

  v2f A;
  A.x = (tv && (kA == k0)) ? (1.0f - w) : ((tv && (kA == k0 + 1)) ? w : 0.0f);
  A.y = (tv && (kA + 1 == k0)) ? (1.0f - w) : ((tv && (kA + 1 == k0 + 1)) ? w : 0.0f);

  // B matrix (4x16 f32): row K striped across lanes; lanes 0-15 K{0,1}, 16-31 K{2,3}
  int n = lane & 15;
  v2f B;
  B.x = 0.0f;
  B.y = 0.0f;
  if (n < NUM_COEFF_) {
#pragma unroll
    for (int r = 0; r < 2; ++r) {
      int K = r + ((lane < 16) ? 0 : 2);
      int fr = base + K;  // <= 99
      float sum = 0.0f, mine = 0.0f;
#pragma unroll
      for (int c = 0; c < NUM_COEFF_; ++c) {
        float sv = sigm(raw_coeff[fr * NUM_COEFF_ + c]);
        sum += sv;
        if (c == n) mine = sv;
      }
      float val = mine / sum * gain;
      if (r == 0) B.x = val; else B.y = val;
    }
  }

  v8f C;
#pragma unroll
  for (int r = 0; r < 8; ++r) C[r] = 0.0f;
  // D = A x B + 0  (exact f32 matrix op)
  v8f D = __builtin_amdgcn_wmma_f32_16x16x4_f32(false, A, false, B, (short)0, C, false, false);

  // D layout: VGPR r -> row r (lanes 0-15) / row r+8 (lanes 16-31), col = lane&15
  int mb = (lane < 16) ? 0 : 8;
  if (n < 8) {
#pragma unroll
    for (int r = 0; r < 8; ++r) d_s[wave][mb + r][n] = D[r];
  }
  __syncthreads();

  // Threads 0..127: one sample each -> vals row
  int i = threadIdx.x;
  if (i < 128) {
    int ts = blockIdx.x * 128 + i;
    if (ts < T_SAMPLES) {
      float b[NUM_COEFF_];
#pragma unroll
      for (int c = 0; c < NUM_COEFF_; ++c) b[c] = d_s[i >> 4][i & 15][c];

      float p = (float)ts * fscale;
      int j0 = (int)floorf(p);
      j0 = j0 < 0 ? 0 : (j0 > 98 ? 98 : j0);
      float wd = p - (float)j0;
      float dlo = delay_frames[j0];
      float dly = dlo + (delay_frames[j0 + 1] - dlo) * wd;
      float zlf = floorf(dly);
      float alfa = dly - zlf;
      float om = 1.0f - alfa;

      float v0 = -om * b[0];
      float v1 = -(alfa * b[0] + om * b[1]);
      float v2 = -(alfa * b[1] + om * b[2]);
      float v3 = -(alfa * b[2] + om * b[3]);
      float v4 = -(alfa * b[3] + om * b[4]);
      float v5 = -(alfa * b[4] + om * b[5]);
      float v6 = -alfa * b[5];

      float4* vp = (float4*)vals;
      vp[ts * 2 + 0] = make_float4(v0, v1, v2, v3);
      vp[ts * 2 + 1] = make_float4(v4, v5, v6, zlf);
    }
  }
}

// ---------------------------------------------------------------------------
// Kernel 2: single persistent workgroup (128 threads = 4 waves).
//  A) async-stage excitation burst + exc coeffs into LDS
//  B) order-5 LPC via 6x6 affine companion block scan (128 blocks x 16 steps)
//  C) main string filter: 437 chunks of 101 samples, history in LDS ring
// ---------------------------------------------------------------------------
__global__ __launch_bounds__(128) void ks_filter(
    const float* __restrict__ excitation,
    const float* __restrict__ exc_coeff,
    const float* __restrict__ vals,
    float* __restrict__ out) {
  __shared__ float xf[BURST];
  __shared__ float ecoef[N_FRAMES_ * EXC_ORDER_];
  __shared__ float blk[128][36];
  __shared__ float ycirc[1024];

  int tid = threadIdx.x;

  // ---- Phase A: stage inputs into LDS ----
#ifdef KS_HAVE_ASYNC
  {
    // param types per hipcc diagnostic: (v4i __device__* , v4i __shared__*, imm, imm)
    const v4i* gsrc = (const v4i*)excitation;
    v4i* ldst = (v4i*)xf;
    for (int ch = tid; ch < (BURST * 4) / 16; ch += 128) {
      __builtin_amdgcn_global_load_async_to_lds_b128(
          (__attribute__((address_space(1))) v4i*)(v4i*)(gsrc + ch),
          (__attribute__((address_space(3))) v4i*)(ldst + ch), 0, 0);
    }
    __builtin_amdgcn_s_wait_asynccnt(0);
  }
#else
  for (int i = tid; i < BURST; i += 128) xf[i] = excitation[i];
#endif
  for (int i = tid; i < N_FRAMES_ * EXC_ORDER_; i += 128) ecoef[i] = exc_coeff[i];
  __syncthreads();

  // ---- Phase B1: per-block 6x6 affine transfer matrices (16 steps each) ----
  {
    float P[6][6];
#pragma unroll
    for (int r = 0; r < 6; ++r)
#pragma unroll
      for (int c = 0; c < 6; ++c) P[r][c] = (r == c) ? 1.0f : 0.0f;

    int t0 = tid * 16;
    for (int s = 0; s < 16; ++s) {
      int t = t0 + s;
      float a[EXC_ORDER_];
      exc_interp(t, ecoef, a);
      float x = xf[t];
      float nr[6];
#pragma unroll
      for (int j = 0; j < 6; ++j) {
        float acc = x * P[5][j];
        acc = fmaf(-a[0], P[0][j], acc);
        acc = fmaf(-a[1], P[1][j], acc);
        acc = fmaf(-a[2], P[2][j], acc);
        acc = fmaf(-a[3], P[3][j], acc);
        acc = fmaf(-a[4], P[4][j], acc);
        nr[j] = acc;
      }
#pragma unroll
      for (int r = 4; r >= 1; --r)
#pragma unroll
        for (int j = 0; j < 6; ++j) P[r][j] = P[r - 1][j];
#pragma unroll
      for (int j = 0; j < 6; ++j) P[0][j] = nr[j];
    }
#pragma unroll
    for (int e = 0; e < 36; ++e) blk[tid][e] = P[e / 6][e % 6];
  }
  __syncthreads();

  // ---- Phase B2: Kogge-Stone inclusive scan of matrix products ----
  for (int d = 1; d < 128; d <<= 1) {
    float L[36], R[36];
    bool act = (tid >= d);
    if (act) {
#pragma unroll
      for (int e = 0; e < 36; ++e) L[e] = blk[tid][e];
#pragma unroll
      for (int e = 0; e < 36; ++e) R[e] = blk[tid - d][e];
    }
    __syncthreads();
    if (act) {
      float M[36];
#pragma unroll
      for (int i2 = 0; i2 < 6; ++i2)
#pragma unroll
        for (int j2 = 0; j2 < 6; ++j2) {
          float acc = 0.0f;
#pragma unroll
          for (int k2 = 0; k2 < 6; ++k2) acc = fmaf(L[i2 * 6 + k2], R[k2 * 6 + j2], acc);
          M[i2 * 6 + j2] = acc;
        }
#pragma unroll
      for (int e = 0; e < 36; ++e) blk[tid][e] = M[e];
    }
    __syncthreads();
  }

  // ---- Phase B3: replay with exact block-start states (column 5 of prefix) ----
  {
    float s0 = 0.0f, s1 = 0.0f, s2 = 0.0f, s3 = 0.0f, s4 = 0.0f;
    if (tid > 0) {
      s0 = blk[tid - 1][0 * 6 + 5];
      s1 = blk[tid - 1][1 * 6 + 5];
      s2 = blk[tid - 1][2 * 6 + 5];
      s3 = blk[tid - 1][3 * 6 + 5];
      s4 = blk[tid - 1][4 * 6 + 5];
    }
    int t0 = tid * 16;
    for (int s = 0; s < 16; ++s) {
      int t = t0 + s;
      float a[EXC_ORDER_];
      exc_interp(t, ecoef, a);
      float x = xf[t];
      float y = x;
      y = fmaf(-a[0], s0, y);
      y = fmaf(-a[1], s1, y);
      y = fmaf(-a[2], s2, y);
      y = fmaf(-a[3], s3, y);
      y = fmaf(-a[4], s4, y);
      s4 = s3; s3 = s2; s2 = s1; s1 = s0; s0 = y;
      xf[t] = y;  // own slot only
    }
  }
  __syncthreads();

  // ---- Phase C: main string filter, chunks of 101 (min lag), ring in LDS ----
  const float4* vp = (const float4*)vals;
  const int nch = (T_SAMPLES + CHUNK - 1) / CHUNK;
  float4 pa = make_float4(0, 0, 0, 0), pb = make_float4(0, 0, 0, 0);
  if (tid < CHUNK && tid < T_SAMPLES) {
    pa = vp[tid * 2 + 0];
    pb = vp[tid * 2 + 1];
  }
  for (int c = 0; c < nch; ++c) {
    int t = c * CHUNK + tid;
    bool act = (tid < CHUNK) && (t < T_SAMPLES);
    float4 a4 = pa, b4 = pb;

    // software prefetch next chunk's tap row (hidden behind compute + barrier)
    int tn = (c + 1) * CHUNK + tid;
    if (tid < CHUNK && tn < T_SAMPLES) {
      pa = vp[tn * 2 + 0];
      pb = vp[tn * 2 + 1];
    }

    if (act) {
      float acc = (t < BURST) ? xf[t] : 0.0f;
      int zl = (int)b4.w;
      int hb = t - zl - 1;  // most recent tap; taps at hb, hb-1, ..., hb-6
      float vv[7] = {a4.x, a4.y, a4.z, a4.w, b4.x, b4.y, b4.z};
#pragma unroll
      for (int jj = 0; jj < 7; ++jj) {
        int h = hb - jj;
        float yh = (h >= 0) ? ycirc[h & 1023] : 0.0f;
        acc = fmaf(-vv[jj], yh, acc);
      }
      // write slot (t mod 1024) held y[t-1024], read window only reaches t-526:
      // no intra-chunk slot aliasing, single barrier per chunk suffices.
      ycirc[t & 1023] = acc;
      out[t] = acc;
    }
    __syncthreads();
  }
}

// ---------------------------------------------------------------------------
extern "C" void kernel_launch(void* const* d_in, const int* in_sizes, int n_in,
                              void* d_out, int out_size, void* d_ws, size_t ws_size,
                              hipStream_t stream) {
  (void)in_sizes; (void)n_in; (void)out_size; (void)ws_size;
  const float* delay_frames = (const float*)d_in[0];
  const float* excitation   = (const float*)d_in[1];
  const float* raw_coeff    = (const float*)d_in[2];
  const float* raw_gain     = (const float*)d_in[3];
  const float* exc_coeffs   = (const float*)d_in[4];
  float* out = (float*)d_out;
  float* vals = (float*)d_ws;  // 44100 * 8 floats = 1.41 MB scratch

  // 345 blocks x 256 threads: 8 waves/block, one 16-sample WMMA tile per wave
  ks_coeff_wmma<<<345, 256, 0, stream>>>(delay_frames, raw_coeff, raw_gain, vals);
  // single persistent workgroup does both sequential filters
  ks_filter<<<1, 128, 0, stream>>>(excitation, exc_coeffs, vals, out);
}